// ExplainableDumplingGNN_80238579024180
// MI455X (gfx1250) — compile-verified
//
#include <hip/hip_runtime.h>
#include <math.h>

#define NN      10000            // nodes
#define EE      160000           // edges
#define HID     64
#define HEADS   8
#define GG      64               // graphs
#define DIN     8
#define HC      512              // HEADS*HID
#define EP      (EE + NN)        // edges incl. self loops
#define KT      64               // GEMM k-panel (all K here are multiples of 64)
#define LDA_S   (KT + 1)         // padded LDS row stride (bank-conflict avoidance)

typedef __attribute__((ext_vector_type(2))) float v2f;
typedef __attribute__((ext_vector_type(8))) float v8f;

// ---------------------------------------------------------------- utilities
__global__ void fill_kernel(float* p, float v, int n) {
  int i = blockIdx.x * blockDim.x + threadIdx.x;
  if (i < n) p[i] = v;
}

__device__ inline void atomicMaxF(float* addr, float val) {
  int* ai = (int*)addr;
  int old = __float_as_int(*addr);
  while (__int_as_float(old) < val) {
    int assumed = old;
    old = atomicCAS(ai, assumed, __float_as_int(val));
    if (old == assumed) break;
  }
}

// ------------------------------------------------- WMMA f32 GEMM (16x16x4)
// C[M,N] = A[M,K] @ B[K,N]   (+= C when accumulate)
// Block: one 16-row M strip, (blockDim/32) N-tiles of 16 cols each.
// A k-panel (16 x KT) staged in LDS with coalesced global loads; fragments
// read back via ds_load. Requirements: M%16==0, N%16==0, K%KT==0.
__global__ void wmma_gemm_f32(const float* __restrict__ A,
                              const float* __restrict__ B,
                              float* __restrict__ C,
                              int M, int Nn, int K, int accumulate) {
  __shared__ float As[16 * LDA_S];

  const int lane = threadIdx.x & 31;
  const int wave = threadIdx.x >> 5;
  const int wavesPerBlk = blockDim.x >> 5;
  const int half = lane >> 4;      // 0: lanes 0-15, 1: lanes 16-31
  const int r    = lane & 15;
  const int tilesN  = Nn >> 4;
  const int groupsN = (tilesN + wavesPerBlk - 1) / wavesPerBlk;
  const int tm = blockIdx.x / groupsN;                        // M tile (strip)
  const int tn = (blockIdx.x % groupsN) * wavesPerBlk + wave; // N tile
  const bool active = (tn < tilesN);                          // wave-uniform

  const float* Bcol = B + (size_t)(active ? tn : 0) * 16 + r;

  v8f c;
  if (active && accumulate) {
#pragma unroll
    for (int v = 0; v < 8; ++v)
      c[v] = C[(size_t)(tm * 16 + v + 8 * half) * Nn + tn * 16 + r];
  } else {
#pragma unroll
    for (int v = 0; v < 8; ++v) c[v] = 0.0f;
  }

  for (int k0 = 0; k0 < K; k0 += KT) {
    // cooperative, coalesced load of the 16 x KT A panel into LDS
    for (int i = threadIdx.x; i < 16 * KT; i += blockDim.x) {
      int row = i >> 6;          // i / KT   (KT == 64)
      int col = i & (KT - 1);    // i % KT
      As[row * LDA_S + col] = A[(size_t)(tm * 16 + row) * K + k0 + col];
    }
    __syncthreads();
    if (active) {
      for (int kk = 0; kk < KT; kk += 4) {
        v2f a, b;
        // A 16x4 frag from LDS: vgpr0 = K{0|2}, vgpr1 = K{1|3} per lane half
        a[0] = As[r * LDA_S + kk + 2 * half + 0];
        a[1] = As[r * LDA_S + kk + 2 * half + 1];
        // B 4x16 frag: rows striped across lanes within a VGPR
        b[0] = Bcol[(size_t)(k0 + kk + 2 * half + 0) * Nn];
        b[1] = Bcol[(size_t)(k0 + kk + 2 * half + 1) * Nn];
        c = __builtin_amdgcn_wmma_f32_16x16x4_f32(false, a, false, b,
                                                  (short)0, c, false, false);
      }
    }
    __syncthreads();
  }

  if (active) {
#pragma unroll
    for (int v = 0; v < 8; ++v)
      C[(size_t)(tm * 16 + v + 8 * half) * Nn + tn * 16 + r] = c[v];
  }
}

// ------------------------------------------------------------- MPNN stage
__global__ void mpnn_lin_kernel(const float* __restrict__ x,
                                const float* __restrict__ w,
                                const float* __restrict__ b,
                                float* __restrict__ xw) {
  int t = blockIdx.x * blockDim.x + threadIdx.x;
  if (t >= NN * HID) return;
  int i = t >> 6, j = t & 63;
  float acc = b[j];
#pragma unroll
  for (int k = 0; k < DIN; ++k) acc += x[i * DIN + k] * w[k * HID + j];
  xw[t] = acc;
}

__global__ void scatter_m_kernel(const int* __restrict__ src,
                                 const int* __restrict__ dst,
                                 const float* __restrict__ xw,
                                 float* __restrict__ m) {
  int t = blockIdx.x * blockDim.x + threadIdx.x;
  if (t >= EE * HID) return;
  int e = t >> 6, j = t & 63;
  atomicAdd(&m[(size_t)dst[e] * HID + j], xw[(size_t)src[e] * HID + j]);
}

__global__ void cnt_kernel(const int* __restrict__ dst, float* __restrict__ cnt) {
  int e = blockIdx.x * blockDim.x + threadIdx.x;
  if (e < EE) atomicAdd(&cnt[dst[e]], 1.0f);
}

__global__ void mpnn_update_kernel(const float* __restrict__ xw,
                                   const float* __restrict__ m,
                                   const float* __restrict__ w,
                                   const float* __restrict__ b,
                                   float* __restrict__ h0) {
  int t = blockIdx.x * blockDim.x + threadIdx.x;
  if (t >= NN * HID) return;
  int i = t >> 6, j = t & 63;
  float acc = b[j];
  for (int k = 0; k < HID; ++k) acc += xw[i * HID + k] * w[k * HID + j];
  for (int k = 0; k < HID; ++k) acc += m[i * HID + k] * w[(HID + k) * HID + j];
  float v = acc > 0.0f ? acc : 0.1f * acc;   // leaky 0.1
  h0[t] = v > 0.0f ? v : 0.0f;               // relu
}

// -------------------------------------------------------------- GAT stage
__global__ void gat_alpha_kernel(const int* __restrict__ src,
                                 const int* __restrict__ dst,
                                 const float* __restrict__ xl,
                                 const float* __restrict__ xr,
                                 const float* __restrict__ att,
                                 float* __restrict__ alpha) {
  int t = blockIdx.x * blockDim.x + threadIdx.x;
  if (t >= EP * HEADS) return;
  int e = t >> 3, h = t & 7;
  int s, d;
  if (e < EE) { s = src[e]; d = dst[e]; } else { s = d = e - EE; }
  const float* pl = xl + (size_t)s * HC + h * HID;
  const float* pr = xr + (size_t)d * HC + h * HID;
  const float* pa = att + h * HID;
  float acc = 0.0f;
  for (int c = 0; c < HID; ++c) {
    float v = pl[c] + pr[c];
    v = v > 0.0f ? v : 0.2f * v;   // leaky 0.2
    acc += v * pa[c];
  }
  alpha[t] = acc;
}

__global__ void gat_amax_kernel(const int* __restrict__ dst,
                                const float* __restrict__ alpha,
                                float* __restrict__ amax) {
  int t = blockIdx.x * blockDim.x + threadIdx.x;
  if (t >= EP * HEADS) return;
  int e = t >> 3, h = t & 7;
  int d = (e < EE) ? dst[e] : (e - EE);
  atomicMaxF(&amax[d * HEADS + h], alpha[t]);
}

__global__ void gat_expsum_kernel(const int* __restrict__ dst,
                                  float* __restrict__ alpha,
                                  const float* __restrict__ amax,
                                  float* __restrict__ denom) {
  int t = blockIdx.x * blockDim.x + threadIdx.x;
  if (t >= EP * HEADS) return;
  int e = t >> 3, h = t & 7;
  int d = (e < EE) ? dst[e] : (e - EE);
  float ea = expf(alpha[t] - amax[d * HEADS + h]);
  alpha[t] = ea;                 // reuse buffer: now holds exp(alpha - amax)
  atomicAdd(&denom[d * HEADS + h], ea);
}

__global__ void gat_scatter_kernel(const int* __restrict__ src,
                                   const int* __restrict__ dst,
                                   const float* __restrict__ xl,
                                   const float* __restrict__ alpha,
                                   const float* __restrict__ denom,
                                   float* __restrict__ acc) {
  int t = blockIdx.x * blockDim.x + threadIdx.x;
  if (t >= EP * HC) return;
  int e = t >> 9, j = t & 511, h = j >> 6;
  int s, d;
  if (e < EE) { s = src[e]; d = dst[e]; } else { s = d = e - EE; }
  float a = alpha[e * HEADS + h] / (denom[d * HEADS + h] + 1e-16f);
  atomicAdd(&acc[(size_t)d * HC + j], xl[(size_t)s * HC + j] * a);
}

__global__ void gat_final_kernel(const float* __restrict__ acc,
                                 const float* __restrict__ bias,
                                 float* __restrict__ h, int act) {
  int t = blockIdx.x * blockDim.x + threadIdx.x;
  if (t >= NN * HC) return;
  int j = t & 511;
  float v = acc[t] + bias[j];
  if (act == 0) v = v > 0.0f ? v : expm1f(v);       // elu
  else          v = v > 0.0f ? v : 0.01f * v;       // leaky 0.01
  h[t] = v;
}

// ------------------------------------------------------------- SAGE stage
__global__ void sage_scatter_kernel(const int* __restrict__ src,
                                    const int* __restrict__ dst,
                                    const float* __restrict__ h,
                                    float* __restrict__ mean) {
  int t = blockIdx.x * blockDim.x + threadIdx.x;
  if (t >= EE * HC) return;
  int e = t >> 9, j = t & 511;
  atomicAdd(&mean[(size_t)dst[e] * HC + j], h[(size_t)src[e] * HC + j]);
}

__global__ void mean_div_kernel(float* __restrict__ mean,
                                const float* __restrict__ cnt) {
  int t = blockIdx.x * blockDim.x + threadIdx.x;
  if (t >= NN * HC) return;
  int i = t >> 9;
  mean[t] /= fmaxf(cnt[i], 1.0f);
}

__global__ void sage_final_kernel(const float* __restrict__ acc,
                                  const float* __restrict__ b,
                                  float* __restrict__ hs) {
  int t = blockIdx.x * blockDim.x + threadIdx.x;
  if (t >= NN * HID) return;
  int j = t & 63;
  float v = acc[t] + b[j];
  hs[t] = v > 0.0f ? v : 0.0f;
}

// ------------------------------------------------------------- pool + head
__global__ void gcnt_kernel(const int* __restrict__ batch, float* __restrict__ gcnt) {
  int i = blockIdx.x * blockDim.x + threadIdx.x;
  if (i < NN) atomicAdd(&gcnt[batch[i]], 1.0f);
}

__global__ void pool_kernel(const int* __restrict__ batch,
                            const float* __restrict__ hs,
                            float* __restrict__ pool) {
  int t = blockIdx.x * blockDim.x + threadIdx.x;
  if (t >= NN * HID) return;
  int i = t >> 6, j = t & 63;
  atomicAdd(&pool[(size_t)batch[i] * HID + j], hs[t]);
}

__global__ void head_kernel(const float* __restrict__ pool,
                            const float* __restrict__ gcnt,
                            const float* __restrict__ ow,
                            const float* __restrict__ ob,
                            float* __restrict__ out) {
  int g = blockIdx.x * blockDim.x + threadIdx.x;
  if (g >= GG) return;
  float inv = 1.0f / fmaxf(gcnt[g], 1.0f);
  float z0 = ob[0], z1 = ob[1];
  for (int j = 0; j < HID; ++j) {
    float p = pool[g * HID + j] * inv;
    z0 += p * ow[j * 2 + 0];
    z1 += p * ow[j * 2 + 1];
  }
  float mx = fmaxf(z0, z1);
  float lse = mx + logf(expf(z0 - mx) + expf(z1 - mx));
  out[g * 2 + 0] = z0 - lse;
  out[g * 2 + 1] = z1 - lse;
}

// --------------------------------------------------------------- launcher
static inline int cdiv(long long a, long long b) { return (int)((a + b - 1) / b); }

static inline void launch_gemm(const float* A, const float* B, float* C,
                               int M, int Nn, int K, int accumulate,
                               int blockThreads, hipStream_t stream) {
  int wavesPerBlk = blockThreads >> 5;
  int tilesM = M >> 4, tilesN = Nn >> 4;
  int groupsN = cdiv(tilesN, wavesPerBlk);
  wmma_gemm_f32<<<tilesM * groupsN, blockThreads, 0, stream>>>(A, B, C, M, Nn, K, accumulate);
}

extern "C" void kernel_launch(void* const* d_in, const int* in_sizes, int n_in,
                              void* d_out, int out_size, void* d_ws, size_t ws_size,
                              hipStream_t stream) {
  const float* x    = (const float*)d_in[0];
  const int*   ei   = (const int*)d_in[1];
  const int*   src  = ei;
  const int*   dst  = ei + EE;
  const int*   batch= (const int*)d_in[2];
  const float* mlw  = (const float*)d_in[3];
  const float* mlb  = (const float*)d_in[4];
  const float* muw  = (const float*)d_in[5];
  const float* mub  = (const float*)d_in[6];
  const float* gwl[3] = {(const float*)d_in[7],  (const float*)d_in[12], (const float*)d_in[17]};
  const float* gwr[3] = {(const float*)d_in[8],  (const float*)d_in[13], (const float*)d_in[18]};
  const float* gat[3] = {(const float*)d_in[9],  (const float*)d_in[14], (const float*)d_in[19]};
  const float* gre[3] = {(const float*)d_in[10], (const float*)d_in[15], (const float*)d_in[20]};
  const float* gbi[3] = {(const float*)d_in[11], (const float*)d_in[16], (const float*)d_in[21]};
  const float* swn  = (const float*)d_in[22];
  const float* sbn  = (const float*)d_in[23];
  const float* swr  = (const float*)d_in[24];
  const float* ow   = (const float*)d_in[25];
  const float* ob   = (const float*)d_in[26];
  float* out = (float*)d_out;

  // ---- workspace layout (floats) ----
  float* W     = (float*)d_ws;
  float* h     = W;                          // NN*HC (first NN*HID also hosts h0)
  float* xl    = h     + (size_t)NN * HC;    // NN*HC (also SAGE mean)
  float* xr    = xl    + (size_t)NN * HC;    // NN*HC
  float* acc   = xr    + (size_t)NN * HC;    // NN*HC (also SAGE acc64)
  float* alpha = acc   + (size_t)NN * HC;    // EP*HEADS
  float* amax  = alpha + (size_t)EP * HEADS; // NN*HEADS
  float* denom = amax  + (size_t)NN * HEADS; // NN*HEADS
  float* xw    = denom + (size_t)NN * HEADS; // NN*HID (also SAGE output hs)
  float* mbuf  = xw    + (size_t)NN * HID;   // NN*HID
  float* cnt   = mbuf  + (size_t)NN * HID;   // NN
  float* gcnt  = cnt   + NN;                 // GG
  float* pool  = gcnt  + GG;                 // GG*HID

  const int B = 256;

  // ---- MPNN ----
  fill_kernel<<<cdiv(NN * HID, B), B, 0, stream>>>(mbuf, 0.0f, NN * HID);
  fill_kernel<<<cdiv(NN, B), B, 0, stream>>>(cnt, 0.0f, NN);
  mpnn_lin_kernel<<<cdiv(NN * HID, B), B, 0, stream>>>(x, mlw, mlb, xw);
  scatter_m_kernel<<<cdiv((long long)EE * HID, B), B, 0, stream>>>(src, dst, xw, mbuf);
  cnt_kernel<<<cdiv(EE, B), B, 0, stream>>>(dst, cnt);
  mpnn_update_kernel<<<cdiv(NN * HID, B), B, 0, stream>>>(xw, mbuf, muw, mub, h);

  // ---- 3 GAT layers ----
  for (int L = 0; L < 3; ++L) {
    const int Kin = (L == 0) ? HID : HC;
    const int act = (L == 1) ? 1 : 0;   // elu, leaky0.01, elu
    launch_gemm(h, gwl[L], xl,  NN, HC, Kin, 0, 256, stream);
    launch_gemm(h, gwr[L], xr,  NN, HC, Kin, 0, 256, stream);
    launch_gemm(h, gre[L], acc, NN, HC, Kin, 0, 256, stream);
    gat_alpha_kernel<<<cdiv((long long)EP * HEADS, B), B, 0, stream>>>(src, dst, xl, xr, gat[L], alpha);
    fill_kernel<<<cdiv(NN * HEADS, B), B, 0, stream>>>(amax, -INFINITY, NN * HEADS);
    gat_amax_kernel<<<cdiv((long long)EP * HEADS, B), B, 0, stream>>>(dst, alpha, amax);
    fill_kernel<<<cdiv(NN * HEADS, B), B, 0, stream>>>(denom, 0.0f, NN * HEADS);
    gat_expsum_kernel<<<cdiv((long long)EP * HEADS, B), B, 0, stream>>>(dst, alpha, amax, denom);
    gat_scatter_kernel<<<cdiv((long long)EP * HC, B), B, 0, stream>>>(src, dst, xl, alpha, denom, acc);
    gat_final_kernel<<<cdiv(NN * HC, B), B, 0, stream>>>(acc, gbi[L], h, act);
  }

  // ---- SAGE (mean aggregation) ----
  fill_kernel<<<cdiv(NN * HC, B), B, 0, stream>>>(xl, 0.0f, NN * HC);   // mean buffer
  sage_scatter_kernel<<<cdiv((long long)EE * HC, B), B, 0, stream>>>(src, dst, h, xl);
  mean_div_kernel<<<cdiv(NN * HC, B), B, 0, stream>>>(xl, cnt);
  launch_gemm(xl, swn, acc, NN, HID, HC, 0, 128, stream);
  launch_gemm(h,  swr, acc, NN, HID, HC, 1, 128, stream);
  sage_final_kernel<<<cdiv(NN * HID, B), B, 0, stream>>>(acc, sbn, xw);  // hs -> xw

  // ---- global mean pool + head ----
  fill_kernel<<<cdiv(GG, B), B, 0, stream>>>(gcnt, 0.0f, GG);
  fill_kernel<<<cdiv(GG * HID, B), B, 0, stream>>>(pool, 0.0f, GG * HID);
  gcnt_kernel<<<cdiv(NN, B), B, 0, stream>>>(batch, gcnt);
  pool_kernel<<<cdiv(NN * HID, B), B, 0, stream>>>(batch, xw, pool);
  head_kernel<<<cdiv(GG, B), B, 0, stream>>>(pool, gcnt, ow, ob, out);
}